// FullyNonlinearKANRNN_17910013624664
// MI455X (gfx1250) — compile-verified
//
#include <hip/hip_runtime.h>
#include <hip/hip_bf16.h>

// FullyNonlinearKANRNN collapsed-recurrence kernel for gfx1250 (MI455X).
//
// The T=16 scan collapses algebraically: final h depends only on x[:,15,:]
// and x[:,14,0]. Head phi(B,640) @ W(640,2->16pad) runs on the WMMA pipe
// (V_WMMA_F32_16X16X4_F32, 160-deep K loop) while the KAN nonlinearity
// runs on the TRANS pipe via the fused identity
//   sigmoid(2*sigmoid(v)) = 0.5 + 0.5*tanh(0.5 + 0.5*tanh(0.5*v))
// = 2x v_tanh_f32 + 1 mul + 2 fma per phi (no divides, no doubling ops).

typedef __attribute__((ext_vector_type(2))) float v2f;
typedef __attribute__((ext_vector_type(8))) float v8f;

#define TT      16
#define DD      3
#define HH      64
#define NBB     10
#define OUTN    2
#define KTOT    (HH * NBB)          // 640
#define ROWS_PER_BLOCK 128          // 8 waves x 16 rows
#define THREADS 256

#if __has_builtin(__builtin_amdgcn_tanhf)
__device__ __forceinline__ float phi_fused(float v) {
    // sigmoid(2*sigmoid(v)) folded through tanh: 2 TRANS + 3 VALU
    const float t1 = __builtin_amdgcn_tanhf(0.5f * v);
    const float u  = __builtin_fmaf(0.5f, t1, 0.5f);     // = (2*sigmoid(v))/2
    const float t2 = __builtin_amdgcn_tanhf(u);
    return __builtin_fmaf(0.5f, t2, 0.5f);
}
#else
__device__ __forceinline__ float fsig(float x) {
    // sigmoid(x) = rcp(1 + 2^(-x*log2e)): v_exp_f32 + v_rcp_f32, no div-fixup
    return __builtin_amdgcn_rcpf(1.0f + __builtin_amdgcn_exp2f(x * -1.4426950408889634f));
}
__device__ __forceinline__ float phi_fused(float v) {
    return fsig(2.0f * fsig(v));
}
#endif

__global__ __launch_bounds__(THREADS)
void kan_rnn_head_kernel(const float* __restrict__ x,
                         const float* __restrict__ a_x,
                         const float* __restrict__ b_x,
                         const float* __restrict__ a_h,
                         const float* __restrict__ b_h,
                         const float* __restrict__ a_o,
                         const float* __restrict__ b_o,
                         const float* __restrict__ W,
                         float* __restrict__ out) {
    __shared__ float sh_h[ROWS_PER_BLOCK * HH];     // 32 KB  collapsed hidden state
    __shared__ float sh_ab[KTOT * 2];               //  5 KB  interleaved {a_o, b_o}
    __shared__ float sh_w[KTOT * OUTN];             //  5 KB  W (640 x 2)
    __shared__ float sh_x[ROWS_PER_BLOCK * 4];      //  2 KB  {x15_0, x15_1, x15_2, x14_0}

    const int tid     = threadIdx.x;
    const int rowBase = blockIdx.x * ROWS_PER_BLOCK;

    // ---- Stage 0: stage head constants + needed x slices into LDS ----
    for (int i = tid; i < KTOT; i += THREADS) {
        sh_ab[2 * i]     = a_o[i];
        sh_ab[2 * i + 1] = b_o[i];
    }
    for (int i = tid; i < KTOT * OUTN; i += THREADS) {
        sh_w[i] = W[i];
    }
    for (int r = tid; r < ROWS_PER_BLOCK; r += THREADS) {
        const float* xr = x + (size_t)(rowBase + r) * (TT * DD);
        const float4 v = *(const float4*)(xr + 44);   // 16B-aligned: {x14_2, x15_0, x15_1, x15_2}
        sh_x[r * 4 + 0] = v.y;
        sh_x[r * 4 + 1] = v.z;
        sh_x[r * 4 + 2] = v.w;
        sh_x[r * 4 + 3] = xr[42];                     // x[row,14,0]
    }
    __syncthreads();

    // ---- Stage 1: closed-form hidden state h[128][64] into LDS ----
    for (int idx = tid; idx < ROWS_PER_BLOCK * HH; idx += THREADS) {
        const int r   = idx >> 6;
        const int col = idx & 63;
        const float* xs = &sh_x[r * 4];
        float h;
        if (col < 30) {
            const int d = (unsigned)col / 10u;                    // 0..2
            h = phi_fused(a_x[col] * (xs[d] - b_x[col]));         // x[row,15,d]
        } else {
            const int k2  = col - 30;                             // 0..33
            const int hh2 = (unsigned)k2 / 10u;                   // 0..3
            const float hp = phi_fused(a_x[hh2] * (xs[3] - b_x[hh2]));
            h = phi_fused(a_h[k2] * (hp - b_h[k2]));
        }
        sh_h[idx] = h;
    }
    __syncthreads();

    // ---- Stage 2: per-wave 16-row tile, phi(16,640) @ W(640,16pad) via WMMA ----
    const int wave = tid >> 5;
    const int lane = tid & 31;
    const int r    = lane & 15;        // tile row (A) == tile col N (B,C)
    const int half = lane >> 4;        // 0: K=c0,c0+1   1: K=c0+2,c0+3
    const int n    = lane & 15;        // output column
    const float* hrow = &sh_h[(wave * 16 + r) * HH];

    // branch-free W access: clamped index + float mask (hoisted out of loop)
    const float wsel = (n < OUTN) ? 1.0f : 0.0f;
    const float* wp  = sh_w + (n & (OUTN - 1));

    v8f acc = {0.f, 0.f, 0.f, 0.f, 0.f, 0.f, 0.f, 0.f};

#pragma unroll 4
    for (int k = 0; k < KTOT / 4; ++k) {
        const int c0 = 4 * k + 2 * half;
        const int c1 = c0 + 1;

        const unsigned hh0 = (unsigned)c0 / 10u;
        const unsigned hh1 = (unsigned)c1 / 10u;
        const v2f ab0 = *(const v2f*)&sh_ab[2 * c0];   // one ds_load_b64: {a_o, b_o}
        const v2f ab1 = *(const v2f*)&sh_ab[2 * c1];

        v2f a;
        a.x = phi_fused(ab0.x * (hrow[hh0] - ab0.y));
        a.y = phi_fused(ab1.x * (hrow[hh1] - ab1.y));

        v2f b;
        b.x = wsel * wp[c0 * OUTN];
        b.y = wsel * wp[c1 * OUTN];

        acc = __builtin_amdgcn_wmma_f32_16x16x4_f32(
            /*neg_a=*/false, a, /*neg_b=*/false, b,
            /*c_mod=*/(short)0, acc, /*reuse_a=*/false, /*reuse_b=*/false);
    }

    // ---- Store: C/D layout => VGPR i : M = i + 8*half, N = lane%16 ----
    if (n < OUTN) {
        const int tileRowBase = rowBase + wave * 16;
#pragma unroll
        for (int i = 0; i < 8; ++i) {
            const int m = i + 8 * half;
            out[(size_t)(tileRowBase + m) * OUTN + n] = acc[i];
        }
    }
}

extern "C" void kernel_launch(void* const* d_in, const int* in_sizes, int n_in,
                              void* d_out, int out_size, void* d_ws, size_t ws_size,
                              hipStream_t stream) {
    (void)n_in; (void)out_size; (void)d_ws; (void)ws_size;
    const float* x   = (const float*)d_in[0];
    const float* a_x = (const float*)d_in[1];
    const float* b_x = (const float*)d_in[2];
    const float* a_h = (const float*)d_in[3];
    const float* b_h = (const float*)d_in[4];
    const float* a_o = (const float*)d_in[5];
    const float* b_o = (const float*)d_in[6];
    const float* W   = (const float*)d_in[7];
    float* out = (float*)d_out;

    const int rows   = in_sizes[0] / (TT * DD);       // 32768
    const int blocks = rows / ROWS_PER_BLOCK;         // 256

    kan_rnn_head_kernel<<<blocks, THREADS, 0, stream>>>(
        x, a_x, b_x, a_h, b_h, a_o, b_o, W, out);
}